// DGCNN_37443524886701
// MI455X (gfx1250) — compile-verified
//
#include <hip/hip_runtime.h>

// ---------------------------------------------------------------------------
// DGCNN forward for MI455X (gfx1250), fp32 WMMA path (V_WMMA_F32_16X16X4_F32).
// Weights pre-packed into K-paired float2 so every fragment is a b64 load;
// register double-buffering hides load latency between v_wmma issues.
// Edge-kernel weight tiles staged via GLOBAL_LOAD_ASYNC_TO_LDS (ASYNCcnt).
// ---------------------------------------------------------------------------

#define NEG_SLOPE 0.2f
#define BATCH 8
#define NPTS 2048
#define KNB 32

typedef __attribute__((ext_vector_type(2))) float v2f;
typedef __attribute__((ext_vector_type(8))) float v8f;

__device__ __forceinline__ v8f wmma4(v2f a, v2f b, v8f c) {
  return __builtin_amdgcn_wmma_f32_16x16x4_f32(false, a, false, b, (short)0, c,
                                               false, false);
}

__device__ __forceinline__ float lrelu(float v) {
  return v >= 0.f ? v : NEG_SLOPE * v;
}

__device__ __forceinline__ v8f vzero8() {
  v8f z = {0.f, 0.f, 0.f, 0.f, 0.f, 0.f, 0.f, 0.f};
  return z;
}

// ---------------------------------------------------------------------------
// Kernel 1: kNN.  One wave per query point; 8 waves / block share pos tile.
// ---------------------------------------------------------------------------
__global__ void knn_kernel(const float* __restrict__ pos, int* __restrict__ idxout) {
  extern __shared__ float smem[];
  float* ppos = smem;                         // [NPTS*3]
  const int lane = threadIdx.x & 31;
  const int wave = threadIdx.x >> 5;          // 0..7
  const int b = blockIdx.x >> 8;
  const int grp = blockIdx.x & 255;
  float* dist = smem + NPTS * 3 + wave * NPTS;

  const float* pb = pos + (size_t)b * NPTS * 3;
  for (int t = threadIdx.x; t < NPTS * 3; t += blockDim.x) ppos[t] = pb[t];
  __syncthreads();

  const int n = grp * 8 + wave;
  const float qx = ppos[n * 3 + 0], qy = ppos[n * 3 + 1], qz = ppos[n * 3 + 2];
  for (int m = lane; m < NPTS; m += 32) {
    float dx = ppos[m * 3 + 0] - qx;
    float dy = ppos[m * 3 + 1] - qy;
    float dz = ppos[m * 3 + 2] - qz;
    dist[m] = dx * dx + dy * dy + dz * dz;
  }
  __syncthreads();

  int* outrow = idxout + ((size_t)b * NPTS + n) * KNB;
  for (int r = 0; r < KNB; ++r) {
    float bv = 3.4e38f;
    int bi = 0x7fffffff;
    for (int m = lane; m < NPTS; m += 32) {
      float v = dist[m];
      if (v < bv) { bv = v; bi = m; }
    }
    #pragma unroll
    for (int off = 16; off > 0; off >>= 1) {
      float ov = __shfl_xor(bv, off, 32);
      int oi = __shfl_xor(bi, off, 32);
      if (ov < bv || (ov == bv && oi < bi)) { bv = ov; bi = oi; }
    }
    if (lane == 0) {
      outrow[r] = b * NPTS + bi;
      dist[bi] = 3.4e38f;
    }
    __syncthreads();
  }
}

// ---------------------------------------------------------------------------
// Kernel 2: pack weights into K-paired float2 (optional subtract for Wdiff)
// ---------------------------------------------------------------------------
__global__ void pack_kernel(const float* __restrict__ B1, const float* __restrict__ B2,
                            v2f* __restrict__ out, int K2, int N) {
  const int t = blockIdx.x * blockDim.x + threadIdx.x;
  if (t >= K2 * N) return;
  const int k2 = t / N, n = t - k2 * N;
  float x0 = B1[(size_t)(2 * k2) * N + n];
  float x1 = B1[(size_t)(2 * k2 + 1) * N + n];
  if (B2) {
    x0 -= B2[(size_t)(2 * k2) * N + n];
    x1 -= B2[(size_t)(2 * k2 + 1) * N + n];
  }
  v2f v; v.x = x0; v.y = x1;
  out[t] = v;
}

// ---------------------------------------------------------------------------
// Kernel 3: fast fp32 WMMA GEMM:  Out = act( A @ B + bias )
// B pre-packed K-paired float2, row stride NN (template -> immediate offsets).
// 32x32 tile per wave (2x2 blocking), register double-buffered K pipeline.
// ---------------------------------------------------------------------------
template <int NN>
__global__ void gemm_fast(const float* __restrict__ A, int lda,
                          const v2f* __restrict__ Bp,
                          const float* __restrict__ bias,
                          float* __restrict__ Out, int ldo,
                          int M, int K, int act) {
  const int lane = threadIdx.x & 31;
  const int wave = threadIdx.x >> 5;
  constexpr int ntn = NN >> 5;
  const int tile = blockIdx.x * (blockDim.x >> 5) + wave;
  const int mt = tile / ntn;
  const int nt = tile - mt * ntn;
  if (mt * 32 >= M) return;                   // wave-uniform exit

  const int half = lane >> 4;
  const int l16 = lane & 15;
  const int row0 = mt * 32 + l16;
  const int col0 = nt * 32 + l16;

  const float* Ap0 = A + (size_t)row0 * lda + 2 * half;
  const float* Ap1 = Ap0 + (size_t)16 * lda;
  const v2f* Bq = Bp + (size_t)half * NN + col0;    // k2 = k0/2 + half

  v8f acc00 = vzero8(), acc01 = vzero8(), acc10 = vzero8(), acc11 = vzero8();

  v2f a0 = *(const v2f*)Ap0;
  v2f a1 = *(const v2f*)Ap1;
  v2f b0 = Bq[0];
  v2f b1 = Bq[16];

  #pragma unroll 4
  for (int k0 = 0; k0 < K - 4; k0 += 4) {
    Ap0 += 4; Ap1 += 4; Bq += 2 * NN;
    v2f na0 = *(const v2f*)Ap0;               // prefetch next step
    v2f na1 = *(const v2f*)Ap1;
    v2f nb0 = Bq[0];
    v2f nb1 = Bq[16];
    acc00 = wmma4(a0, b0, acc00);
    acc01 = wmma4(a0, b1, acc01);
    acc10 = wmma4(a1, b0, acc10);
    acc11 = wmma4(a1, b1, acc11);
    a0 = na0; a1 = na1; b0 = nb0; b1 = nb1;
  }
  acc00 = wmma4(a0, b0, acc00);
  acc01 = wmma4(a0, b1, acc01);
  acc10 = wmma4(a1, b0, acc10);
  acc11 = wmma4(a1, b1, acc11);

  const float bv0 = bias ? bias[col0] : 0.f;
  const float bv1 = bias ? bias[col0 + 16] : 0.f;
  float* O0 = Out + (size_t)(mt * 32 + 8 * half) * ldo + col0;
  #pragma unroll
  for (int r = 0; r < 8; ++r) {
    float v00 = acc00[r] + bv0;
    float v01 = acc01[r] + bv1;
    float v10 = acc10[r] + bv0;
    float v11 = acc11[r] + bv1;
    if (act) { v00 = lrelu(v00); v01 = lrelu(v01); v10 = lrelu(v10); v11 = lrelu(v11); }
    float* Or = O0 + (size_t)r * ldo;
    Or[0] = v00;
    Or[16] = v01;
    Or[(size_t)16 * ldo] = v10;
    Or[(size_t)16 * ldo + 16] = v11;
  }
}

// ---------------------------------------------------------------------------
// Kernel 4: naive GEMM for odd shapes (K=3 pre-block1, N=12 head3).
// ---------------------------------------------------------------------------
__global__ void gemm_naive(const float* __restrict__ A, int lda,
                           const float* __restrict__ B1, const float* __restrict__ B2,
                           int ldb, const float* __restrict__ bias,
                           float* __restrict__ Out, int ldo,
                           int M, int N, int K, int act) {
  const int t = blockIdx.x * blockDim.x + threadIdx.x;
  if (t >= M * N) return;
  const int row = t / N;
  const int col = t - row * N;
  float s = bias ? bias[col] : 0.f;
  const float* Ar = A + (size_t)row * lda;
  for (int k = 0; k < K; ++k) {
    float w = B1[(size_t)k * ldb + col];
    if (B2) w -= B2[(size_t)k * ldb + col];
    s += Ar[k] * w;
  }
  if (act) s = lrelu(s);
  Out[(size_t)row * ldo + col] = s;
}

// ---------------------------------------------------------------------------
// Kernel 5: edge block second GEMM + maxpool (the dominant GEMM).
//   h1[n,j,:] = lrelu( Aall[n,:] + Gall[idx[n,j],:] )        (built in LDS)
//   out[n,:]  = lrelu( max_j (h1[n,j,:] @ Wb) + bb )         (WMMA + reg max)
// One wave per point (2 row-tiles), 4 col-tiles per pass (8 wmma / K-step).
// Weight tile staged K-paired into LDS via GLOBAL_LOAD_ASYNC_TO_LDS_B32
// (per-lane LDS destination, SGPR base + 32-bit voffset), sync on ASYNCcnt.
// ---------------------------------------------------------------------------
template <int CA>
__global__ void edge_kernel(const float* __restrict__ Aall,
                            const float* __restrict__ Gall,
                            const int* __restrict__ idx,
                            const float* __restrict__ Wb,
                            const float* __restrict__ bb,
                            float* __restrict__ Out,
                            int Cb, int ldo) {
  extern __shared__ float smem[];
  constexpr int HP = CA + 4;                  // padded LDS row stride (even)
  constexpr int AV = CA / 64;                 // v2f per lane per h1 row
  const int lane = threadIdx.x & 31;
  const int wave = threadIdx.x >> 5;
  const int P = blockDim.x >> 5;              // points per block (=4)
  float* H1 = smem;                           // [P][32][HP]
  float* Btf = smem + P * KNB * HP;           // [CA/2][64] float2 pairs
  const v2f* Bt2 = (const v2f*)Btf;

  const int gn = blockIdx.x * P + wave;       // global point row
  float* Hp = H1 + (size_t)wave * KNB * HP;

  // ---- phase a: gather + first-layer activation into LDS (v2f traffic) ----
  const int myg = idx[(size_t)gn * KNB + lane];
  const float* Arow = Aall + (size_t)gn * CA;
  v2f apre[AV];
  #pragma unroll
  for (int i = 0; i < AV; ++i) apre[i] = *(const v2f*)&Arow[2 * lane + 64 * i];
  for (int j = 0; j < KNB; ++j) {
    const int g = __shfl(myg, j, 32);
    const float* Grow = Gall + (size_t)g * CA;
    #pragma unroll
    for (int i = 0; i < AV; ++i) {
      v2f gv = *(const v2f*)&Grow[2 * lane + 64 * i];
      v2f hv;
      hv.x = lrelu(apre[i].x + gv.x);
      hv.y = lrelu(apre[i].y + gv.y);
      *(v2f*)&Hp[j * HP + 2 * lane + 64 * i] = hv;
    }
  }
  __syncthreads();

  // ---- phase b: WMMA over groups of 4 column tiles + maxpool ---------------
  const int half = lane >> 4;
  const int l16 = lane & 15;
  const int nct = Cb >> 4;                    // 16-col tiles (multiple of 4)
  for (int nt = 0; nt < nct; nt += 4) {
    // stage Wb[:, nt*16 .. nt*16+63] K-paired into LDS, async (no VGPR hop).
    // CA*64 is a multiple of blockDim (128) -> EXEC full on every issue.
    for (int t = threadIdx.x; t < CA * 64; t += blockDim.x) {
      const int k = t >> 6, c = t & 63;
      const unsigned ldst =
          (unsigned)(size_t)&Btf[(k >> 1) * 128 + c * 2 + (k & 1)];
      const unsigned goff =
          ((unsigned)k * (unsigned)Cb + (unsigned)(nt * 16 + c)) * 4u;
      asm volatile("global_load_async_to_lds_b32 %0, %1, %2"
                   :: "v"(ldst), "v"(goff), "s"(Wb)
                   : "memory");
    }
    asm volatile("s_wait_asynccnt 0x0" ::: "memory");
    __syncthreads();

    v8f acc0[4] = {vzero8(), vzero8(), vzero8(), vzero8()};
    v8f acc1[4] = {vzero8(), vzero8(), vzero8(), vzero8()};
    #pragma unroll 4
    for (int k0 = 0; k0 < CA; k0 += 4) {
      const int kk = k0 + 2 * half;
      const int kk2 = (k0 >> 1) + half;
      v2f a0 = *(const v2f*)&Hp[l16 * HP + kk];
      v2f a1 = *(const v2f*)&Hp[(16 + l16) * HP + kk];
      #pragma unroll
      for (int ct = 0; ct < 4; ++ct) {
        v2f b = Bt2[kk2 * 64 + ct * 16 + l16];
        acc0[ct] = wmma4(a0, b, acc0[ct]);
        acc1[ct] = wmma4(a1, b, acc1[ct]);
      }
    }

    // maxpool over the 32 edge rows (bias row-constant; lrelu monotone)
    #pragma unroll
    for (int ct = 0; ct < 4; ++ct) {
      float m = acc0[ct][0];
      #pragma unroll
      for (int r = 1; r < 8; ++r) m = fmaxf(m, acc0[ct][r]);
      #pragma unroll
      for (int r = 0; r < 8; ++r) m = fmaxf(m, acc1[ct][r]);
      m = fmaxf(m, __shfl_xor(m, 16, 32));
      if (lane < 16) {
        const int col = (nt + ct) * 16 + l16;
        Out[(size_t)gn * ldo + col] = lrelu(m + bb[col]);
      }
    }
    __syncthreads();
  }
}

// ---------------------------------------------------------------------------
// Host orchestration
// ---------------------------------------------------------------------------
template <int NN>
static inline void launch_fast(const float* A, int lda, const v2f* Bp,
                               const float* bias, float* Out, int ldo,
                               int M, int K, int act, hipStream_t s) {
  const int tiles = (M / 32) * (NN / 32);
  const int blocks = (tiles + 3) / 4;
  gemm_fast<NN><<<blocks, 128, 0, s>>>(A, lda, Bp, bias, Out, ldo, M, K, act);
}

extern "C" void kernel_launch(void* const* d_in, const int* in_sizes, int n_in,
                              void* d_out, int out_size, void* d_ws, size_t ws_size,
                              hipStream_t stream) {
  (void)in_sizes; (void)n_in; (void)out_size; (void)ws_size;

  const float* x    = (const float*)d_in[0];
  const float* pos  = (const float*)d_in[1];
  const float* w1a  = (const float*)d_in[2];
  const float* b1a  = (const float*)d_in[3];
  const float* w1b  = (const float*)d_in[4];
  const float* b1b  = (const float*)d_in[5];
  const float* w2a  = (const float*)d_in[6];
  const float* b2a  = (const float*)d_in[7];
  const float* w2b  = (const float*)d_in[8];
  const float* b2b  = (const float*)d_in[9];
  const float* w3a  = (const float*)d_in[10];
  const float* b3a  = (const float*)d_in[11];
  const float* w3b  = (const float*)d_in[12];
  const float* b3b  = (const float*)d_in[13];
  const float* wf1  = (const float*)d_in[14];
  const float* bf1  = (const float*)d_in[15];
  const float* wf2  = (const float*)d_in[16];
  const float* bf2  = (const float*)d_in[17];
  const float* wf3  = (const float*)d_in[18];
  const float* bf3  = (const float*)d_in[19];
  float* out = (float*)d_out;

  const int M = BATCH * NPTS;                 // 16384 rows

  // workspace layout (~118 MB)
  char* ws = (char*)d_ws;
  size_t off = 0;
  int*   idx  = (int*)(ws + off);   off += (size_t)M * KNB * 4;       // 2 MB
  float* Aall = (float*)(ws + off); off += (size_t)M * 128 * 4;       // 8 MB
  float* Gall = (float*)(ws + off); off += (size_t)M * 128 * 4;       // 8 MB
  float* hcat = (float*)(ws + off); off += (size_t)M * 704 * 4;       // 46 MB
  float* hh1  = (float*)(ws + off); off += (size_t)M * 512 * 4;       // 33.5 MB
  float* hh2  = (float*)(ws + off); off += (size_t)M * 256 * 4;       // 16 MB
  v2f* wd2p = (v2f*)(ws + off); off += (size_t)32 * 64 * 8;           // packed Wdiff2
  v2f* wg2p = (v2f*)(ws + off); off += (size_t)32 * 64 * 8;           // packed W2a_bot
  v2f* wd3p = (v2f*)(ws + off); off += (size_t)64 * 128 * 8;          // packed Wdiff3
  v2f* wg3p = (v2f*)(ws + off); off += (size_t)64 * 128 * 8;          // packed W3a_bot
  v2f* wf1p = (v2f*)(ws + off); off += (size_t)352 * 512 * 8;         // packed wf1
  v2f* wf2p = (v2f*)(ws + off); off += (size_t)256 * 256 * 8;         // packed wf2

  // allow >64KB dynamic LDS
  const int knn_smem = (NPTS * 3 + 8 * NPTS) * 4;                     // 88 KB
  (void)hipFuncSetAttribute((const void*)knn_kernel,
                            hipFuncAttributeMaxDynamicSharedMemorySize, knn_smem);
  const int edge_smem64 = (4 * KNB * (64 + 4) + 64 * 64) * 4;         // 50 KB
  const int edge_smem128 = (4 * KNB * (128 + 4) + 128 * 64) * 4;      // 98 KB
  (void)hipFuncSetAttribute((const void*)edge_kernel<64>,
                            hipFuncAttributeMaxDynamicSharedMemorySize, edge_smem64);
  (void)hipFuncSetAttribute((const void*)edge_kernel<128>,
                            hipFuncAttributeMaxDynamicSharedMemorySize, edge_smem128);

  // 1) kNN graph (idx holds global rows b*N+m)
  knn_kernel<<<BATCH * 256, 256, knn_smem, stream>>>(pos, idx);

  // weight packing (Wdiff = Wa_top - Wa_bot fused where needed)
  pack_kernel<<<(32 * 64 + 255) / 256, 256, 0, stream>>>(w2a, w2a + 64 * 64, wd2p, 32, 64);
  pack_kernel<<<(32 * 64 + 255) / 256, 256, 0, stream>>>(w2a + 64 * 64, nullptr, wg2p, 32, 64);
  pack_kernel<<<(64 * 128 + 255) / 256, 256, 0, stream>>>(w3a, w3a + 128 * 128, wd3p, 64, 128);
  pack_kernel<<<(64 * 128 + 255) / 256, 256, 0, stream>>>(w3a + 128 * 128, nullptr, wg3p, 64, 128);
  pack_kernel<<<(352 * 512 + 255) / 256, 256, 0, stream>>>(wf1, nullptr, wf1p, 352, 512);
  pack_kernel<<<(256 * 256 + 255) / 256, 256, 0, stream>>>(wf2, nullptr, wf2p, 256, 256);

  // 2) block1: Cin=3 -> Ca=64 -> Cb=64   (K=3 -> naive pre-GEMMs)
  gemm_naive<<<(M * 64 + 255) / 256, 256, 0, stream>>>(
      x, 3, w1a, w1a + 3 * 64, 64, b1a, Aall, 64, M, 64, 3, 0);
  gemm_naive<<<(M * 64 + 255) / 256, 256, 0, stream>>>(
      x, 3, w1a + 3 * 64, nullptr, 64, nullptr, Gall, 64, M, 64, 3, 0);
  edge_kernel<64><<<M / 4, 128, edge_smem64, stream>>>(Aall, Gall, idx, w1b, b1b,
                                                       hcat + 0, 64, 704);

  // 3) block2: Cin=64 (x1 = hcat[:, 0:64]) -> Ca=64 -> Cb=128
  launch_fast<64>(hcat, 704, wd2p, b2a, Aall, 64, M, 64, 0, stream);
  launch_fast<64>(hcat, 704, wg2p, nullptr, Gall, 64, M, 64, 0, stream);
  edge_kernel<64><<<M / 4, 128, edge_smem64, stream>>>(Aall, Gall, idx, w2b, b2b,
                                                       hcat + 64, 128, 704);

  // 4) block3: Cin=128 (x2 = hcat[:, 64:192]) -> Ca=128 -> Cb=512
  launch_fast<128>(hcat + 64, 704, wd3p, b3a, Aall, 128, M, 128, 0, stream);
  launch_fast<128>(hcat + 64, 704, wg3p, nullptr, Gall, 128, M, 128, 0, stream);
  edge_kernel<128><<<M / 4, 128, edge_smem128, stream>>>(Aall, Gall, idx, w3b, b3b,
                                                         hcat + 192, 512, 704);

  // 5) head: 704 -> 512 -> 256 -> 12
  launch_fast<512>(hcat, 704, wf1p, bf1, hh1, 512, M, 704, 1, stream);
  launch_fast<256>(hh1, 512, wf2p, bf2, hh2, 256, M, 512, 1, stream);
  gemm_naive<<<(M * 12 + 255) / 256, 256, 0, stream>>>(
      hh2, 256, wf3, nullptr, 12, bf3, out, 12, M, 12, 256, 0);
}